// Basicgate_patch_iv_multivoxel_66726611910723
// MI455X (gfx1250) — compile-verified
//
#include <hip/hip_runtime.h>
#include <hip/hip_bf16.h>

// ---------------- problem constants (match reference) ----------------
#define H_    192
#define W_    640
#define HW_   (H_ * W_)          // 122880
#define BATCH 2
#define MTOT  (BATCH * HW_)      // 245760 pixels
#define CIMG  64
#define C0    32                 // feat0 channels
#define CL    67                 // C1 + 3
#define C0P   36                 // 35 padded to mult of 4
#define CLP   68                 // 67 padded to mult of 4
#define NT    80                 // 67 out-channels padded to 5 N-tiles of 16
#define TPAD  16                 // 9 conv taps padded to one N-tile

typedef __attribute__((ext_vector_type(2))) float v2f;
typedef __attribute__((ext_vector_type(8))) float v8f;

// ---------------- scatter: last-write-wins via atomicMax on point idx ---
__global__ void scatter_idx_kernel(const float* __restrict__ coor,
                                   int* __restrict__ idx, int N) {
    int t = blockIdx.x * blockDim.x + threadIdx.x;
    int b = blockIdx.y;
    if (t >= N) return;
    const float* c = coor + ((size_t)b * N + t) * 2;
    float u = fminf(fmaxf(c[0], 0.f), 1.f);
    float v = fminf(fmaxf(c[1], 0.f), 1.f);
    int row = (int)(v * (float)H_);   // coor[:, ::-1] * (H, W), truncated
    int col = (int)(u * (float)W_);
    if (row >= H_ || col >= W_) return;      // cropped overflow row/col
    atomicMax(&idx[b * HW_ + row * W_ + col], t);
}

// gather winning point's (feat ++ vox) into channel-last padded grid
__global__ void gather_kernel(const int* __restrict__ idx,
                              const float* __restrict__ feat,
                              const float* __restrict__ vox,
                              float* __restrict__ grid,
                              int N, int Cfeat, int Cpad) {
    int C = Cfeat + 3;
    size_t gid = (size_t)blockIdx.x * blockDim.x + threadIdx.x;
    size_t total = (size_t)MTOT * C;
    if (gid >= total) return;
    int cell = (int)(gid / C);
    int c    = (int)(gid % C);
    int i = idx[cell];
    if (i < 0) return;                        // grid pre-zeroed
    int b = cell / HW_;
    float val = (c < Cfeat) ? feat[((size_t)b * N + i) * Cfeat + c]
                            : vox[((size_t)b * N + i) * 3 + (c - Cfeat)];
    grid[(size_t)cell * Cpad + c] = val;
}

// ---------------- per-pixel gate: 1x1 conv of x_rgb0 down to 1 channel --
__global__ void gated_kernel(const float* __restrict__ x,
                             const float* __restrict__ w3,
                             const float* __restrict__ b3,
                             float* __restrict__ gated) {
    int p = blockIdx.x * blockDim.x + threadIdx.x;
    if (p >= MTOT) return;
    int b = p / HW_, hw = p % HW_;
    const float* xb = x + (size_t)b * CIMG * HW_ + hw;
    float acc = b3[0];
#pragma unroll
    for (int c = 0; c < CIMG; ++c) acc += xb[(size_t)c * HW_] * w3[c];
    gated[p] = acc;
}

// ------- fused chained 1x1 convs + conv3x3 tap dots, all via WMMA -------
//   E1    = W0 * p0 + b0 + p1
//   fusedv= W2 * E1 + b2 + gated          (kept in LDS only)
//   dotp[p][tap] = fusedv[p] . W_sp[:,tap]  (tap = ky*3+kx, 9 taps)
__global__ __launch_bounds__(128)
void gemm_fused_kernel(const float* __restrict__ grid0,   // [MTOT][36]
                       const float* __restrict__ grid1,   // [MTOT][68]
                       const float* __restrict__ gated,   // [MTOT]
                       const float* __restrict__ W_red0,  // [67][35]
                       const float* __restrict__ b_red0,  // [67]
                       const float* __restrict__ W_red2,  // [67][67]
                       const float* __restrict__ b_red2,  // [67]
                       const float* __restrict__ W_sp,    // [67][3][3]
                       float* __restrict__ dotp)          // [MTOT][16]
{
    // weights in [n][k] order -> every B fetch is one aligned ds_load_b64
    __shared__ float sW0[NT * C0P];          // W_red0, zero-padded
    __shared__ float sW2[NT * CLP];          // W_red2, zero-padded
    __shared__ float sWs[TPAD * CLP];        // W_sp taps, zero-padded
    __shared__ float sA0[4][16 * C0P];       // per-wave p0 tile
    __shared__ float sA1[4][16 * CLP];       // per-wave E1 tile, then fused tile

    const int tid = threadIdx.x;
    for (int i = tid; i < NT * C0P; i += 128) {
        int n = i / C0P, k = i % C0P;
        sW0[i] = (n < CL && k < 35) ? W_red0[n * 35 + k] : 0.f;
    }
    for (int i = tid; i < NT * CLP; i += 128) {
        int n = i / CLP, k = i % CLP;
        sW2[i] = (n < CL && k < CL) ? W_red2[n * CL + k] : 0.f;
    }
    for (int i = tid; i < TPAD * CLP; i += 128) {
        int n = i / CLP, k = i % CLP;
        sWs[i] = (n < 9 && k < CL) ? W_sp[k * 9 + n] : 0.f;
    }

    const int wave = tid >> 5;
    const int lane = tid & 31;
    const int gm0 = (blockIdx.x * 4 + wave) * 16;   // first pixel of wave tile
    float* A0 = sA0[wave];
    float* A1 = sA1[wave];

    // 16 consecutive channel-last rows == one contiguous 576-float block
    const float* src0 = grid0 + (size_t)gm0 * C0P;
    for (int i = lane; i < 16 * C0P; i += 32) A0[i] = src0[i];
    if (lane < 16) A1[lane * CLP + CL] = 0.f;       // K-pad column stays zero
    __syncthreads();

    const int mrow  = lane & 15;             // A layout: M across lanes 0-15
    const int khalf = (lane >> 4) * 2;       // lanes 16-31 carry K+2,K+3
    const int mAcc  = (lane >> 4) * 8;       // C layout: M = vgpr + 8*(lane>=16)

    // -------- GEMM 1: E1 = W0 * p0  (K=36, 9 steps, 5 N-tiles) -----------
    v2f a0r[9];
#pragma unroll
    for (int s = 0; s < 9; ++s)
        a0r[s] = *(const v2f*)&A0[mrow * C0P + 4 * s + khalf];

#pragma unroll
    for (int t = 0; t < 5; ++t) {
        v8f acc = {};
        const int nb = t * 16 + mrow;        // B/C layout: N = lane&15
#pragma unroll
        for (int s = 0; s < 9; ++s) {
            v2f b = *(const v2f*)&sW0[nb * C0P + 4 * s + khalf];
            acc = __builtin_amdgcn_wmma_f32_16x16x4_f32(
                false, a0r[s], false, b, (short)0, acc, false, false);
        }
        if (nb < CL) {
            float bias0 = b_red0[nb];
#pragma unroll
            for (int j = 0; j < 8; ++j) {
                int m = j + mAcc;
                float p1 = grid1[(size_t)(gm0 + m) * CLP + nb];
                A1[m * CLP + nb] = acc[j] + bias0 + p1;
            }
        }
    }
    __syncthreads();

    // -------- GEMM 2: E2 = W2 * E1  (K=68, 17 steps, 5 N-tiles) ----------
    v2f a1r[17];
#pragma unroll
    for (int s = 0; s < 17; ++s)
        a1r[s] = *(const v2f*)&A1[mrow * CLP + 4 * s + khalf];

    float e2buf[5][8];                        // buffer so A1 isn't clobbered
#pragma unroll
    for (int t = 0; t < 5; ++t) {
        v8f acc = {};
        const int nb = t * 16 + mrow;
#pragma unroll
        for (int s = 0; s < 17; ++s) {
            v2f b = *(const v2f*)&sW2[nb * CLP + 4 * s + khalf];
            acc = __builtin_amdgcn_wmma_f32_16x16x4_f32(
                false, a1r[s], false, b, (short)0, acc, false, false);
        }
#pragma unroll
        for (int j = 0; j < 8; ++j) e2buf[t][j] = acc[j];
    }
    __syncthreads();

    // restage fused = E2 + b2 + gated into A1 (A-layout) for the tap GEMM
    float gv[8];
#pragma unroll
    for (int j = 0; j < 8; ++j) gv[j] = gated[gm0 + j + mAcc];
#pragma unroll
    for (int t = 0; t < 5; ++t) {
        const int nb = t * 16 + mrow;
        if (nb < CL) {
            float bias2 = b_red2[nb];
#pragma unroll
            for (int j = 0; j < 8; ++j)
                A1[(j + mAcc) * CLP + nb] = e2buf[t][j] + bias2 + gv[j];
        }
    }
    __syncthreads();

    // -------- tap GEMM: dotp = fused * W_sp taps  (K=68, 17 steps, N=9) --
    {
#pragma unroll
        for (int s = 0; s < 17; ++s)
            a1r[s] = *(const v2f*)&A1[mrow * CLP + 4 * s + khalf];
        v8f acc = {};
#pragma unroll
        for (int s = 0; s < 17; ++s) {
            v2f b = *(const v2f*)&sWs[mrow * CLP + 4 * s + khalf];
            acc = __builtin_amdgcn_wmma_f32_16x16x4_f32(
                false, a1r[s], false, b, (short)0, acc, false, false);
        }
        if (mrow < 9) {
#pragma unroll
            for (int j = 0; j < 8; ++j)
                dotp[(size_t)(gm0 + j + mAcc) * TPAD + mrow] = acc[j];
        }
    }
}

// ------- attention: sum 9 neighbor taps, sigmoid, broadcast multiply ----
__global__ void att_out_kernel(const float* __restrict__ dotp,  // [MTOT][16]
                               const float* __restrict__ bsp,
                               const float* __restrict__ x,
                               float* __restrict__ out) {
    int p = blockIdx.x * blockDim.x + threadIdx.x;
    if (p >= MTOT) return;
    int b = p / HW_, hw = p % HW_;
    int h = hw / W_, w = hw % W_;
    float acc = bsp[0];
#pragma unroll
    for (int ky = 0; ky < 3; ++ky) {
        int hh = h + ky - 1;
        if (hh < 0 || hh >= H_) continue;
#pragma unroll
        for (int kx = 0; kx < 3; ++kx) {
            int ww = w + kx - 1;
            if (ww < 0 || ww >= W_) continue;
            acc += dotp[(size_t)(b * HW_ + hh * W_ + ww) * TPAD + (ky * 3 + kx)];
        }
    }
    float att = 1.f / (1.f + __expf(-acc));
    const float* xb = x + (size_t)b * CIMG * HW_ + hw;
    float* ob = out + (size_t)b * CIMG * HW_ + hw;
#pragma unroll
    for (int c = 0; c < CIMG; ++c) ob[(size_t)c * HW_] = xb[(size_t)c * HW_] * att;
}

// ------------------------------ launcher --------------------------------
extern "C" void kernel_launch(void* const* d_in, const int* in_sizes, int n_in,
                              void* d_out, int out_size, void* d_ws, size_t ws_size,
                              hipStream_t stream) {
    const float* x_rgb0 = (const float*)d_in[0];
    const float* feat0  = (const float*)d_in[1];
    const float* coor0  = (const float*)d_in[2];
    const float* vox0   = (const float*)d_in[3];
    const float* feat1  = (const float*)d_in[4];
    const float* coor1  = (const float*)d_in[5];
    const float* vox1   = (const float*)d_in[6];
    const float* W_red0 = (const float*)d_in[7];
    const float* b_red0 = (const float*)d_in[8];
    const float* W_red2 = (const float*)d_in[9];
    const float* b_red2 = (const float*)d_in[10];
    const float* W_red3 = (const float*)d_in[11];
    const float* b_red3 = (const float*)d_in[12];
    const float* W_sp   = (const float*)d_in[13];
    const float* b_sp   = (const float*)d_in[14];
    float* out = (float*)d_out;

    const int N0 = in_sizes[2] / (BATCH * 2);   // coor0 is [B,N0,2]
    const int N1 = in_sizes[5] / (BATCH * 2);   // coor1 is [B,N1,2]

    // workspace carve-up (floats)
    const size_t G0 = (size_t)MTOT * C0P;       // grid0
    const size_t G1 = (size_t)MTOT * CLP;       // grid1
    float* grid0 = (float*)d_ws;
    float* grid1 = grid0 + G0;
    float* dotp  = grid1 + G1;                  // MTOT*16
    float* gated = dotp + (size_t)MTOT * TPAD;
    int*   idx0  = (int*)(gated + MTOT);
    int*   idx1  = idx0 + MTOT;

    // zero grids (pad channels stay zero -> free WMMA K-padding), idx = -1
    hipMemsetAsync(grid0, 0, (G0 + G1) * sizeof(float), stream);
    hipMemsetAsync(idx0, 0xFF, 2 * (size_t)MTOT * sizeof(int), stream);

    // scatter (last-write-wins via atomicMax on point index)
    {
        dim3 g0((N0 + 255) / 256, BATCH), g1((N1 + 255) / 256, BATCH);
        scatter_idx_kernel<<<g0, 256, 0, stream>>>(coor0, idx0, N0);
        scatter_idx_kernel<<<g1, 256, 0, stream>>>(coor1, idx1, N1);
    }
    {
        size_t t0 = (size_t)MTOT * (C0 + 3);
        size_t t1 = (size_t)MTOT * CL;
        gather_kernel<<<(int)((t0 + 255) / 256), 256, 0, stream>>>(
            idx0, feat0, vox0, grid0, N0, C0, C0P);
        gather_kernel<<<(int)((t1 + 255) / 256), 256, 0, stream>>>(
            idx1, feat1, vox1, grid1, N1, CL - 3, CLP);
    }

    gated_kernel<<<(MTOT + 255) / 256, 256, 0, stream>>>(x_rgb0, W_red3, b_red3, gated);

    // WMMA GEMM: 15360 M-tiles of 16 pixels, 4 waves (tiles) per block
    gemm_fused_kernel<<<MTOT / 16 / 4, 128, 0, stream>>>(
        grid0, grid1, gated, W_red0, b_red0, W_red2, b_red2, W_sp, dotp);

    att_out_kernel<<<(MTOT + 255) / 256, 256, 0, stream>>>(dotp, b_sp, x_rgb0, out);
}